// OneWinnerMHNLayer_4879082848728
// MI455X (gfx1250) — compile-verified
//
#include <hip/hip_runtime.h>
#include <hip/hip_bf16.h>
#include <stdint.h>

// ---------------------------------------------------------------------------
// OneWinnerMHN on MI455X (gfx1250, wave32).
// B=32, S=513 (T=512 ctx + 1 query), D=1024, K=256, V=256, H=2048, O=256.
//
// Restructured: winner updates are full row replacements, so
//   logits_t[h] = last[h]<0 ? L0[t,h] : G[last[h],t]
// with L0 = xK @ W_hi0^T and G = xK @ xK^T as dense batched GEMMs (WMMA f32),
// and the sequential scan reduced to argmax bookkeeping over last[],
// with the L0 row double-buffered in LDS via async-to-LDS loads.
// ---------------------------------------------------------------------------

typedef __attribute__((ext_vector_type(2))) float v2f;
typedef __attribute__((ext_vector_type(8))) float v8f;

#define B_   32
#define T_   512
#define D_   1024
#define K_   256
#define V_   256
#define H_   2048
#define O_   256

// ---------------------------------------------------------------------------
// Generic batched NT GEMM via V_WMMA_F32_16X16X4_F32:
//   C[b, m, n] = sum_k A[b, m, k] * Bt[b, n, k]
// One wave computes a 64(M) x 64(N) tile: 4 A-frags x 4 B-frags -> 16 accs.
// Per k-step: 8 x global_load_b64, 16 x wmma  (16 F/B from cache, 16
// independent accumulation chains -> WMMA-issue bound).
// A-frag layout (16x4 f32): lane<16 -> row=lane,  k = kk + {0,1}
//                           lane>=16 -> row=lane-16, k = kk + {2,3}
// B-frag layout (4x16 f32): mirror of A with col=lane&15.
// C layout: acc[j] -> row +j+8*(lane>=16), col +(lane&15).
// ---------------------------------------------------------------------------
__global__ void __launch_bounds__(32)
gemm_nt_wmma(const float* __restrict__ A, long long strideA, int lda,
             const float* __restrict__ Bt, long long strideB, int ldb,
             float* __restrict__ C, long long strideC, int ldc,
             int Kdim) {
  const int b    = blockIdx.z;
  const int m0   = blockIdx.y * 64;
  const int n0   = blockIdx.x * 64;
  const int lane = threadIdx.x & 31;
  const int half = lane >> 4;     // 0: K pair {0,1}, 1: K pair {2,3}
  const int lr   = lane & 15;

  const float* a_base = A  + (long long)b * strideA + (long long)(m0 + lr) * lda + 2 * half;
  const float* b_base = Bt + (long long)b * strideB + (long long)(n0 + lr) * ldb + 2 * half;

  v8f acc[4][4];
  #pragma unroll
  for (int mi = 0; mi < 4; ++mi)
    #pragma unroll
    for (int ni = 0; ni < 4; ++ni)
      acc[mi][ni] = (v8f){};

  for (int kk = 0; kk < Kdim; kk += 4) {
    v2f af[4], bf[4];
    #pragma unroll
    for (int i = 0; i < 4; ++i)
      af[i] = *(const v2f*)(a_base + (long long)(16 * i) * lda + kk);
    #pragma unroll
    for (int i = 0; i < 4; ++i)
      bf[i] = *(const v2f*)(b_base + (long long)(16 * i) * ldb + kk);
    #pragma unroll
    for (int mi = 0; mi < 4; ++mi)
      #pragma unroll
      for (int ni = 0; ni < 4; ++ni)
        acc[mi][ni] = __builtin_amdgcn_wmma_f32_16x16x4_f32(
            false, af[mi], false, bf[ni], (short)0, acc[mi][ni], false, false);
  }

  float* c_base = C + (long long)b * strideC;
  #pragma unroll
  for (int mi = 0; mi < 4; ++mi) {
    #pragma unroll
    for (int j = 0; j < 8; ++j) {
      const long long row = (long long)(m0 + 16 * mi + j + 8 * half) * ldc;
      #pragma unroll
      for (int ni = 0; ni < 4; ++ni)
        c_base[row + n0 + 16 * ni + lr] = acc[mi][ni][j];
    }
  }
}

// ---------------------------------------------------------------------------
// xQ[b, n] = dot(x[b, T_, :], W_Q[n, :])   (tiny: 32x256x1024)
// ---------------------------------------------------------------------------
__global__ void __launch_bounds__(256)
xq_kernel(const float* __restrict__ x, const float* __restrict__ W_Q,
          float* __restrict__ xQ) {
  const int b = blockIdx.x, n = threadIdx.x;
  __shared__ float xr[D_];
  const float* xrow = x + ((long long)b * (T_ + 1) + T_) * D_;
  for (int i = n; i < D_; i += 256) xr[i] = xrow[i];
  __syncthreads();
  const float* w = W_Q + (long long)n * D_;
  float acc = 0.f;
  for (int k = 0; k < D_; ++k) acc += xr[k] * w[k];
  xQ[b * K_ + n] = acc;
}

// ---------------------------------------------------------------------------
// Sequential winner scan. One block per batch, 1024 threads (32 waves).
// Per step t: value[h] = last[h]<0 ? L0row[h] : G[last[h],t]; argmax -> winner;
// last[winner] = t.  Tie-break: first (lowest h) max, matching jnp.argmax.
// The L0 row for step t+1 is prefetched into LDS with async-to-LDS loads
// (ASYNCcnt) while the argmax reduction for step t runs.
// ---------------------------------------------------------------------------
__global__ void __launch_bounds__(1024)
scan_kernel(const float* __restrict__ L0,   // [B, T, H]
            const float* __restrict__ G,    // [B, T, T]
            int* __restrict__ last_out) {   // [B, H]
  const int b = blockIdx.x;
  const int tid = threadIdx.x;
  __shared__ __align__(16) float rowbuf[2][H_];   // 16 KB double buffer
  __shared__ int   last[H_];
  __shared__ float sval[1024];
  __shared__ int   sidx[1024];

  last[tid] = -1;
  last[tid + 1024] = -1;

  const float* L0b = L0 + (long long)b * T_ * H_;
  const float* Gb  = G  + (long long)b * T_ * T_;

  // issue async load of row t into rowbuf[t&1]: each thread moves 8 bytes
  auto issue_row = [&](int t) {
    const float* g = L0b + (long long)t * H_ + 2 * tid;
    uint32_t lds = (uint32_t)(uintptr_t)(&rowbuf[t & 1][2 * tid]);
    asm volatile("global_load_async_to_lds_b64 %0, %1, off"
                 :: "v"(lds), "v"(g) : "memory");
  };

  issue_row(0);

  for (int t = 0; t < T_; ++t) {
    asm volatile("s_wait_asynccnt 0x0" ::: "memory");
    __syncthreads();                       // row t visible to all waves
    if (t + 1 < T_) issue_row(t + 1);      // overlap with reduction below

    const float* row = rowbuf[t & 1];
    int   l0 = last[tid];
    int   l1 = last[tid + 1024];
    float v0 = (l0 < 0) ? row[tid]        : Gb[(long long)l0 * T_ + t];
    float v1 = (l1 < 0) ? row[tid + 1024] : Gb[(long long)l1 * T_ + t];
    float bestv = v0; int besti = tid;
    if (v1 > bestv) { bestv = v1; besti = tid + 1024; }
    sval[tid] = bestv; sidx[tid] = besti;
    __syncthreads();
    for (int s = 512; s > 0; s >>= 1) {
      if (tid < s) {
        float ov = sval[tid + s]; int oi = sidx[tid + s];
        if (ov > sval[tid] || (ov == sval[tid] && oi < sidx[tid])) {
          sval[tid] = ov; sidx[tid] = oi;
        }
      }
      __syncthreads();
    }
    if (tid == 0) last[sidx[0]] = t;
    __syncthreads();
  }

  last_out[b * H_ + tid]        = last[tid];
  last_out[b * H_ + tid + 1024] = last[tid + 1024];
}

// ---------------------------------------------------------------------------
// p[b,h] = softmax_h( dot(W_hi_final[b,h], xQ[b]) ),
// W_hi_final row = last<0 ? W_hi0 row : xK[b, last].
// ---------------------------------------------------------------------------
__global__ void __launch_bounds__(256)
softmax_p_kernel(const float* __restrict__ W_hi0, const float* __restrict__ xK,
                 const float* __restrict__ xQ, const int* __restrict__ last,
                 float* __restrict__ p) {
  const int b = blockIdx.x, tid = threadIdx.x;
  __shared__ float q[K_];
  __shared__ float red[256];
  q[tid] = xQ[b * K_ + tid];
  __syncthreads();

  float s[8];
  float mx = -INFINITY;
  #pragma unroll
  for (int i = 0; i < 8; ++i) {
    const int h = tid + i * 256;
    const int l = last[b * H_ + h];
    const float* row = (l < 0) ? (W_hi0 + ((long long)b * H_ + h) * K_)
                               : (xK    + ((long long)b * T_ + l) * K_);
    float acc = 0.f;
    for (int k = 0; k < K_; ++k) acc += row[k] * q[k];
    s[i] = acc;                    // BETA = 1.0
    mx = fmaxf(mx, acc);
  }
  red[tid] = mx; __syncthreads();
  for (int st = 128; st > 0; st >>= 1) {
    if (tid < st) red[tid] = fmaxf(red[tid], red[tid + st]);
    __syncthreads();
  }
  const float bmax = red[0];
  __syncthreads();
  float lsum = 0.f;
  #pragma unroll
  for (int i = 0; i < 8; ++i) { s[i] = expf(s[i] - bmax); lsum += s[i]; }
  red[tid] = lsum; __syncthreads();
  for (int st = 128; st > 0; st >>= 1) {
    if (tid < st) red[tid] += red[tid + st];
    __syncthreads();
  }
  const float inv = 1.f / red[0];
  #pragma unroll
  for (int i = 0; i < 8; ++i) p[b * H_ + tid + i * 256] = s[i] * inv;
}

// ---------------------------------------------------------------------------
// mhn_out[v] = sum_h p[h] * (last[h]<0 ? W_oh0[b,v,h] : xV[b,last[h],v]);
// out = mhn_out @ W_proj^T.  One block per batch, 256 threads.
// ---------------------------------------------------------------------------
__global__ void __launch_bounds__(256)
mhn_out_kernel(const float* __restrict__ W_oh0, const float* __restrict__ xV,
               const float* __restrict__ W_proj, const int* __restrict__ last,
               const float* __restrict__ p, float* __restrict__ out) {
  const int b = blockIdx.x, v = threadIdx.x;
  __shared__ float pl[H_];
  __shared__ int   ll[H_];
  __shared__ float mo[V_];
  for (int i = v; i < H_; i += 256) { pl[i] = p[b * H_ + i]; ll[i] = last[b * H_ + i]; }
  __syncthreads();

  const float* Wb  = W_oh0 + ((long long)b * V_ + v) * H_;
  const float* xVb = xV + (long long)b * T_ * V_;
  float acc = 0.f;
  for (int h = 0; h < H_; ++h) {
    const int l = ll[h];
    const float w = (l < 0) ? Wb[h] : xVb[(long long)l * V_ + v];
    acc += pl[h] * w;
  }
  mo[v] = acc;
  __syncthreads();

  const float* Wp = W_proj + (long long)v * V_;   // thread index = output o
  float oacc = 0.f;
  for (int k = 0; k < V_; ++k) oacc += Wp[k] * mo[k];
  out[b * O_ + v] = oacc;
}

// ---------------------------------------------------------------------------
// reinst[b,d] = sum_h p[h] * (last[h]<0 ? W_reinst0[b,h,d] : x[b,last[h],d])
// ---------------------------------------------------------------------------
__global__ void __launch_bounds__(1024)
reinst_kernel(const float* __restrict__ W_reinst0, const float* __restrict__ x,
              const int* __restrict__ last, const float* __restrict__ p,
              float* __restrict__ reinst) {
  const int b = blockIdx.x, d = threadIdx.x;
  __shared__ float pl[H_];
  __shared__ int   ll[H_];
  for (int i = d; i < H_; i += 1024) { pl[i] = p[b * H_ + i]; ll[i] = last[b * H_ + i]; }
  __syncthreads();

  const float* Wb = W_reinst0 + (long long)b * H_ * D_;
  const float* xb = x + (long long)b * (T_ + 1) * D_;
  float acc = 0.f;
  for (int h = 0; h < H_; ++h) {
    const int l = ll[h];
    const float w = (l < 0) ? Wb[(long long)h * D_ + d] : xb[(long long)l * D_ + d];
    acc += pl[h] * w;
  }
  reinst[b * D_ + d] = acc;
}

// ---------------------------------------------------------------------------
extern "C" void kernel_launch(void* const* d_in, const int* in_sizes, int n_in,
                              void* d_out, int out_size, void* d_ws, size_t ws_size,
                              hipStream_t stream) {
  const float* x         = (const float*)d_in[0];
  const float* W_Q       = (const float*)d_in[1];
  const float* W_K       = (const float*)d_in[2];
  const float* W_V       = (const float*)d_in[3];
  const float* W_proj    = (const float*)d_in[4];
  const float* W_hi0     = (const float*)d_in[5];
  const float* W_oh0     = (const float*)d_in[6];
  const float* W_reinst0 = (const float*)d_in[7];

  float* out    = (float*)d_out;              // [B, O]
  float* reinst = out + (size_t)B_ * O_;      // [B, D]
  float* xQ     = reinst + (size_t)B_ * D_;   // [B, K] (also an output)

  float* ws = (float*)d_ws;
  size_t o = 0;
  float* xK = ws + o; o += (size_t)B_ * T_ * K_;   // 4.19M
  float* xV = ws + o; o += (size_t)B_ * T_ * V_;   // 4.19M
  float* L0 = ws + o; o += (size_t)B_ * T_ * H_;   // 33.6M
  float* G  = ws + o; o += (size_t)B_ * T_ * T_;   // 8.39M
  float* p  = ws + o; o += (size_t)B_ * H_;        // 65K
  int* last = (int*)(ws + o);                      // [B, H] ints

  // 1) xQ (tiny GEMV)
  xq_kernel<<<dim3(B_), dim3(256), 0, stream>>>(x, W_Q, xQ);

  // 2) xK = ctx @ W_K^T ; xV = ctx @ W_V^T   (A: x rows 0..511 of each batch)
  dim3 gProj(K_ / 64, T_ / 64, B_);
  gemm_nt_wmma<<<gProj, dim3(32), 0, stream>>>(
      x, (long long)(T_ + 1) * D_, D_, W_K, 0LL, D_, xK, (long long)T_ * K_, K_, D_);
  gemm_nt_wmma<<<gProj, dim3(32), 0, stream>>>(
      x, (long long)(T_ + 1) * D_, D_, W_V, 0LL, D_, xV, (long long)T_ * V_, V_, D_);

  // 3) L0 = xK @ W_hi0^T  (512 x 2048 per batch)
  dim3 gL0(H_ / 64, T_ / 64, B_);
  gemm_nt_wmma<<<gL0, dim3(32), 0, stream>>>(
      xK, (long long)T_ * K_, K_, W_hi0, (long long)H_ * K_, K_,
      L0, (long long)T_ * H_, H_, K_);

  // 4) G = xK @ xK^T  (Gram, 512 x 512 per batch)
  dim3 gG(T_ / 64, T_ / 64, B_);
  gemm_nt_wmma<<<gG, dim3(32), 0, stream>>>(
      xK, (long long)T_ * K_, K_, xK, (long long)T_ * K_, K_,
      G, (long long)T_ * T_, T_, K_);

  // 5) sequential winner scan (async-to-LDS double-buffered L0 rows)
  scan_kernel<<<dim3(B_), dim3(1024), 0, stream>>>(L0, G, last);

  // 6) softmax over reconstructed W_hi @ xQ
  softmax_p_kernel<<<dim3(B_), dim3(256), 0, stream>>>(W_hi0, xK, xQ, last, p);

  // 7) read-outs
  mhn_out_kernel<<<dim3(B_), dim3(256), 0, stream>>>(W_oh0, xV, W_proj, last, p, out);
  reinst_kernel<<<dim3(B_), dim3(1024), 0, stream>>>(W_reinst0, x, last, p, reinst);
}